// DSTGCN_50096498540654
// MI455X (gfx1250) — compile-verified
//
#include <hip/hip_runtime.h>

#define B_ 8
#define T_ 12
#define N_ 800
#define D_ 16
#define CO_ 32
#define KS_ 3
#define LN_EPS 1e-5f

typedef __attribute__((ext_vector_type(2))) float v2f;
typedef __attribute__((ext_vector_type(8))) float v8f;

// ---------------------------------------------------------------------------
// Kernel 1/3: Out[bt] = G[b] @ In[bt]
//   G  : (B, N, N) row-major
//   In : (B*T, N, D) row-major
//   Out: (B*T, N, D) row-major
// One wave computes one 16x16 output tile (16 rows x D=16), K-loop over N=800
// in steps of 4 using V_WMMA_F32_16X16X4_F32.
// A layout (ISA 7.12.2, 32-bit A 16x4): lanes 0-15 hold (M=lane, K=0/1),
// lanes 16-31 hold (M=lane-16, K=2/3).
// ---------------------------------------------------------------------------
__global__ __launch_bounds__(64) void gemm_gx(const float* __restrict__ G,
                                              const float* __restrict__ In,
                                              float* __restrict__ Out) {
  const int wave   = threadIdx.x >> 5;
  const int lane   = threadIdx.x & 31;
  const int lane16 = lane & 15;
  const int khalf  = lane >> 4;            // 0 -> K=0,1 ; 1 -> K=2,3
  const int mtile  = blockIdx.x * 2 + wave; // 0..49
  const int bt     = blockIdx.y;            // 0..B*T-1
  const int b      = bt / T_;
  const int mbase  = mtile * 16;

  const float* Gp = G + (size_t)b * N_ * N_ + (size_t)(mbase + lane16) * N_;
  const float* Xp = In + (size_t)bt * N_ * D_;

  v8f c = {};
#pragma unroll 4
  for (int k0 = 0; k0 < N_; k0 += 4) {
    const int ka = k0 + 2 * khalf;
    v2f a = *(const v2f*)(Gp + ka);          // 8B-aligned contiguous pair
    v2f bf;
    bf.x = Xp[ka * D_ + lane16];
    bf.y = Xp[(ka + 1) * D_ + lane16];
    c = __builtin_amdgcn_wmma_f32_16x16x4_f32(false, a, false, bf,
                                              (short)0, c, false, false);
  }

  float* Op = Out + (size_t)bt * N_ * D_;
#pragma unroll
  for (int r = 0; r < 8; ++r) {
    // C layout: VGPR r = row (r + 8*khalf), col = lane16
    Op[(size_t)(mbase + r + 8 * khalf) * D_ + lane16] = c[r];
  }
}

// ---------------------------------------------------------------------------
// Kernel 2: X1[b,t,n,d] = p12[n]*X[b,t-1,n,d] + Y[b,t,n,d] + p32[n]*X[b,t+1,n,d]
// (time indices edge-clamped, matching the x_pad replication)
// ---------------------------------------------------------------------------
__global__ void mix_x1(const float* __restrict__ X, const float* __restrict__ Y,
                       const float* __restrict__ p12, const float* __restrict__ p32,
                       float* __restrict__ X1) {
  const int idx = blockIdx.x * blockDim.x + threadIdx.x;
  const int total = B_ * T_ * N_ * D_;
  if (idx >= total) return;
  const int n = (idx / D_) % N_;
  const int t = (idx / (D_ * N_)) % T_;
  const int b = idx / (D_ * N_ * T_);
  const int tm = t > 0 ? t - 1 : 0;
  const int tp = t < T_ - 1 ? t + 1 : T_ - 1;
  const size_t nd = (size_t)N_ * D_;
  const size_t off = idx - (size_t)((b * T_ + t)) * nd;  // n*D + d
  const float xm = X[((size_t)b * T_ + tm) * nd + off];
  const float xp = X[((size_t)b * T_ + tp) * nd + off];
  X1[idx] = p12[n] * xm + Y[idx] + p32[n] * xp;
}

// ---------------------------------------------------------------------------
// Kernel 4: h = X@W0 + X1@W1 + X2@W2 (K=48, CO=32) then LayerNorm over CO.
// X2 built on the fly:
//   x1_top = Y(t-1) + p21*X(t); x1_bot = p23*X(t) + Y(t+1)
//   X2 = 2*(p12*x1_top + Z + p32*x1_bot) - X(t)
// One wave per 16-row tile; two f32 WMMA accumulators (CO cols 0-15, 16-31).
// ---------------------------------------------------------------------------
__global__ __launch_bounds__(32) void final_ln(
    const float* __restrict__ X, const float* __restrict__ Y,
    const float* __restrict__ X1, const float* __restrict__ Z,
    const float* __restrict__ W,   // (KS, D, CO)
    const float* __restrict__ p12, const float* __restrict__ p21,
    const float* __restrict__ p23, const float* __restrict__ p32,
    const float* __restrict__ gamma, const float* __restrict__ beta,
    float* __restrict__ out) {
  const int lane   = threadIdx.x & 31;
  const int lane16 = lane & 15;
  const int khalf  = lane >> 4;
  const int mtile  = blockIdx.x;  // 0..49
  const int bt     = blockIdx.y;
  const int t  = bt % T_;
  const int b  = bt / T_;
  const int tm = t > 0 ? t - 1 : 0;
  const int tp = t < T_ - 1 ? t + 1 : T_ - 1;
  const int mbase = mtile * 16;
  const int row   = mbase + lane16;   // A-side row this lane supplies

  const size_t nd = (size_t)N_ * D_;
  const float* Xt  = X  + ((size_t)b * T_ + t)  * nd + (size_t)row * D_;
  const float* Yp  = Y  + ((size_t)b * T_ + tm) * nd + (size_t)row * D_;
  const float* Yn  = Y  + ((size_t)b * T_ + tp) * nd + (size_t)row * D_;
  const float* X1t = X1 + (size_t)bt * nd + (size_t)row * D_;
  const float* Zt  = Z  + (size_t)bt * nd + (size_t)row * D_;

  const float q12 = p12[row], q21 = p21[row];
  const float q23 = p23[row], q32 = p32[row];

  v8f c0 = {}, c1 = {};
#pragma unroll
  for (int chunk = 0; chunk < KS_; ++chunk) {
    const float* Wk = W + (size_t)chunk * D_ * CO_;
#pragma unroll
    for (int k0 = 0; k0 < D_; k0 += 4) {
      const int ka = k0 + 2 * khalf;
      v2f a;
      if (chunk == 0) {
        a.x = Xt[ka];
        a.y = Xt[ka + 1];
      } else if (chunk == 1) {
        a.x = X1t[ka];
        a.y = X1t[ka + 1];
      } else {
        const float xt0 = Xt[ka],      xt1 = Xt[ka + 1];
        const float top0 = Yp[ka]     + q21 * xt0;
        const float top1 = Yp[ka + 1] + q21 * xt1;
        const float bot0 = q23 * xt0 + Yn[ka];
        const float bot1 = q23 * xt1 + Yn[ka + 1];
        a.x = 2.0f * (q12 * top0 + Zt[ka]     + q32 * bot0) - xt0;
        a.y = 2.0f * (q12 * top1 + Zt[ka + 1] + q32 * bot1) - xt1;
      }
      v2f b0, b1;
      b0.x = Wk[ka * CO_ + lane16];
      b0.y = Wk[(ka + 1) * CO_ + lane16];
      b1.x = Wk[ka * CO_ + 16 + lane16];
      b1.y = Wk[(ka + 1) * CO_ + 16 + lane16];
      c0 = __builtin_amdgcn_wmma_f32_16x16x4_f32(false, a, false, b0,
                                                 (short)0, c0, false, false);
      c1 = __builtin_amdgcn_wmma_f32_16x16x4_f32(false, a, false, b1,
                                                 (short)0, c1, false, false);
    }
  }

  // LayerNorm over CO=32: row m's 32 values live in c0[m%8], c1[m%8] across
  // one 16-lane half-wave; reduce with xor masks 1,2,4,8 (stays within half).
  const float g0 = gamma[lane16], g1 = gamma[16 + lane16];
  const float be0 = beta[lane16], be1 = beta[16 + lane16];
  float* Op = out + (size_t)bt * N_ * CO_;
#pragma unroll
  for (int r = 0; r < 8; ++r) {
    const float v0 = c0[r], v1 = c1[r];
    float s  = v0 + v1;
    float s2 = v0 * v0 + v1 * v1;
#pragma unroll
    for (int m = 1; m < 16; m <<= 1) {
      s  += __shfl_xor(s, m, 32);
      s2 += __shfl_xor(s2, m, 32);
    }
    const float mu  = s * (1.0f / CO_);
    const float var = s2 * (1.0f / CO_) - mu * mu;
    const float inv = rsqrtf(var + LN_EPS);
    const int orow = mbase + r + 8 * khalf;
    Op[(size_t)orow * CO_ + lane16]      = (v0 - mu) * inv * g0 + be0;
    Op[(size_t)orow * CO_ + 16 + lane16] = (v1 - mu) * inv * g1 + be1;
  }
}

extern "C" void kernel_launch(void* const* d_in, const int* in_sizes, int n_in,
                              void* d_out, int out_size, void* d_ws, size_t ws_size,
                              hipStream_t stream) {
  const float* x     = (const float*)d_in[0];  // (B,T,N,D)
  const float* gso   = (const float*)d_in[1];  // (B,N,N)
  const float* w     = (const float*)d_in[2];  // (KS,D,CO)
  const float* p12   = (const float*)d_in[3];
  const float* p21   = (const float*)d_in[4];
  const float* p23   = (const float*)d_in[5];
  const float* p32   = (const float*)d_in[6];
  const float* gamma = (const float*)d_in[7];
  const float* beta  = (const float*)d_in[8];
  float* out = (float*)d_out;

  const size_t slab = (size_t)B_ * T_ * N_ * D_;  // 1,228,800 floats
  float* Y  = (float*)d_ws;
  float* X1 = Y + slab;
  float* Z  = X1 + slab;

  dim3 gGemm(25, B_ * T_);                         // 2 waves/block x 25 = 50 tiles
  gemm_gx<<<gGemm, 64, 0, stream>>>(gso, x, Y);

  const int total = B_ * T_ * N_ * D_;
  mix_x1<<<(total + 255) / 256, 256, 0, stream>>>(x, Y, p12, p32, X1);

  gemm_gx<<<gGemm, 64, 0, stream>>>(gso, X1, Z);

  dim3 gFin(50, B_ * T_);
  final_ln<<<gFin, 32, 0, stream>>>(x, Y, X1, Z, w, p12, p21, p23, p32,
                                    gamma, beta, out);
}